// MeanProbExtractor_yolov2_14353780703310
// MI455X (gfx1250) — compile-verified
//
#include <hip/hip_runtime.h>
#include <math.h>

// Problem constants (YOLOv2 COCO, 26x26 grid, 5 anchors)
#define NBOX   3380          // 5 * 26 * 26
#define GRID   26
#define CELLS  676           // 26*26
#define CHAN   85            // 5 + 80 classes
#define CONF_T 0.5f
#define IOU_T  0.4f
#define TPB    1024
#define OWN    4             // ceil(NBOX / TPB)

typedef float v2f __attribute__((ext_vector_type(2)));
typedef float v8f __attribute__((ext_vector_type(8)));

__constant__ float ANCH_W[5] = {0.57273f, 1.87446f, 3.33843f, 7.88282f, 9.77052f};
__constant__ float ANCH_H[5] = {0.677385f, 2.06253f, 5.47434f, 3.52778f, 9.16828f};

extern "C" __global__ __launch_bounds__(TPB)
void yolo2_nms_meanconf_kernel(const float* __restrict__ in, float* __restrict__ out)
{
    // Dynamic LDS: 7 arrays of NBOX floats/ints = 94,640 bytes (CDNA5 WGP has 320KB)
    extern __shared__ float smem[];
    float* s_uconf = smem;                 // unsorted conf
    float* s_conf  = smem + 1 * NBOX;      // sorted conf
    float* s_cx    = smem + 2 * NBOX;
    float* s_cy    = smem + 3 * NBOX;
    float* s_w     = smem + 4 * NBOX;
    float* s_h     = smem + 5 * NBOX;
    int*   s_sup   = (int*)(smem + 6 * NBOX);   // keep LDS addrspace -> ds_load/ds_store

    const int tid = threadIdx.x;

    float lcx[OWN], lcy[OWN], lw[OWN], lh[OWN], lconf[OWN];
    int   lidx[OWN];
    int   nown = 0;

    // -------- Phase 0: decode batch-0 boxes (only image 0 enters NMS) --------
    for (int n = tid; n < NBOX; n += TPB) {
        int a  = n / CELLS;
        int r  = n - a * CELLS;
        int gy = r / GRID;
        int gx = r - gy * GRID;
        const float* p = in + (size_t)(a * CHAN) * CELLS + r;
        __builtin_prefetch(p, 0, 3);               // global_prefetch_b8
        float tx = p[0];
        float ty = p[1 * CELLS];
        float tw = p[2 * CELLS];
        float th = p[3 * CELLS];
        float to = p[4 * CELLS];
        float sx = 1.0f / (1.0f + expf(-tx));
        float sy = 1.0f / (1.0f + expf(-ty));
        float so = 1.0f / (1.0f + expf(-to));
        float cx = (sx + (float)gx) / 26.0f;       // match reference op order
        float cy = (sy + (float)gy) / 26.0f;
        float w  = expf(tw) * ANCH_W[a] / 26.0f;
        float h  = expf(th) * ANCH_H[a] / 26.0f;

        s_uconf[n]  = so;
        lcx[nown]   = cx;  lcy[nown] = cy;
        lw[nown]    = w;   lh[nown]  = h;
        lconf[nown] = so;  lidx[nown] = n;
        ++nown;
    }
    __syncthreads();

    // -------- Phase 1: stable descending rank (== argsort(-conf)) + scatter --------
    {
        int rk[OWN];
        for (int m = 0; m < OWN; ++m) rk[m] = 0;
        for (int j = 0; j < NBOX; ++j) {
            float cj = s_uconf[j];                 // LDS broadcast read
            for (int m = 0; m < nown; ++m) {
                float ci = lconf[m];
                rk[m] += (int)((cj > ci) || ((cj == ci) && (j < lidx[m])));
            }
        }
        for (int m = 0; m < nown; ++m) {
            int p = rk[m];
            s_conf[p] = lconf[m];
            s_cx[p]   = lcx[m];  s_cy[p] = lcy[m];
            s_w[p]    = lw[m];   s_h[p]  = lh[m];
            s_sup[p]  = 0;
        }
    }
    __syncthreads();

    // -------- Phase 2: greedy NMS (decision is workgroup-uniform; barrier only on keep) ----
    for (int i = 0; i < NBOX; ++i) {
        float ci  = s_conf[i];
        bool kept = (ci > CONF_T) && (s_sup[i] == 0);
        if (kept) {
            float cxi = s_cx[i], cyi = s_cy[i], wi = s_w[i], hi = s_h[i];
            float x1i = cxi - 0.5f * wi, x2i = cxi + 0.5f * wi;
            float y1i = cyi - 0.5f * hi, y2i = cyi + 0.5f * hi;
            float ai  = wi * hi;
            for (int j = i + 1 + tid; j < NBOX; j += TPB) {
                float cxj = s_cx[j], cyj = s_cy[j], wj = s_w[j], hj = s_h[j];
                float x1j = cxj - 0.5f * wj, x2j = cxj + 0.5f * wj;
                float y1j = cyj - 0.5f * hj, y2j = cyj + 0.5f * hj;
                float uw  = fmaxf(x2i, x2j) - fminf(x1i, x1j);
                float uh  = fmaxf(y2i, y2j) - fminf(y1i, y1j);
                float cw  = wi + wj - uw;
                float ch  = hi + hj - uh;
                float inter = ((cw <= 0.0f) || (ch <= 0.0f)) ? 0.0f : cw * ch;
                float uni   = ai + wj * hj - inter;
                float iou   = inter / uni;         // match reference exactly
                if (iou > IOU_T) s_sup[j] = 1;     // same-value writes: race-free
            }
            __syncthreads();
        }
    }
    __syncthreads();

    // -------- Phase 3: WMMA f32 reduction of (sum conf*keep, count keep) on wave 0 ----
    if (tid < 32) {                                // full wave32 -> EXEC all-ones for WMMA
        v8f accC = {0.f, 0.f, 0.f, 0.f, 0.f, 0.f, 0.f, 0.f};
        v8f accK = {0.f, 0.f, 0.f, 0.f, 0.f, 0.f, 0.f, 0.f};
        v2f bones = {1.0f, 1.0f};                  // B = ones(4x16) -> D[m,n] += sum_k A[m,k]

        const int lane  = tid;
        const int low   = lane & 15;
        const int hiofs = (lane >> 4) << 5;        // lanes 16-31 hold K=2,3 halves
        const int nch   = (NBOX + 63) / 64;

        for (int c = 0; c < nch; ++c) {
            int base = c * 64 + hiofs + low;
            v2f ac, ak;
            {
                int idx = base;                    // A VGPR0 element
                float cv = 0.f, kv = 0.f;
                if (idx < NBOX) {
                    float cc = s_conf[idx];
                    if ((cc > CONF_T) && (s_sup[idx] == 0)) { cv = cc; kv = 1.0f; }
                }
                ac.x = cv; ak.x = kv;
            }
            {
                int idx = base + 16;               // A VGPR1 element
                float cv = 0.f, kv = 0.f;
                if (idx < NBOX) {
                    float cc = s_conf[idx];
                    if ((cc > CONF_T) && (s_sup[idx] == 0)) { cv = cc; kv = 1.0f; }
                }
                ac.y = cv; ak.y = kv;
            }
            accC = __builtin_amdgcn_wmma_f32_16x16x4_f32(
                false, ac, false, bones, (short)0, accC, false, false);
            accK = __builtin_amdgcn_wmma_f32_16x16x4_f32(
                false, ak, false, bones, (short)0, accK, false, false);
        }

        float tC = 0.f, tK = 0.f;
        for (int r = 0; r < 8; ++r) { tC += accC[r]; tK += accK[r]; }
        // column n of D lives split across lane n (M=0..7) and lane n+16 (M=8..15)
        tC += __shfl_xor(tC, 16, 32);
        tK += __shfl_xor(tK, 16, 32);

        if (lane == 0) {
            out[0] = (tK > 0.5f) ? (tC / tK) : 0.0f;
        }
    }
}

extern "C" void kernel_launch(void* const* d_in, const int* in_sizes, int n_in,
                              void* d_out, int out_size, void* d_ws, size_t ws_size,
                              hipStream_t stream) {
    (void)in_sizes; (void)n_in; (void)d_ws; (void)ws_size; (void)out_size;
    const float* in  = (const float*)d_in[0];   // [64, 425, 26, 26] f32
    float*       out = (float*)d_out;           // scalar f32
    const size_t lds_bytes = (size_t)7 * NBOX * sizeof(float);  // 94,640 B
    yolo2_nms_meanconf_kernel<<<dim3(1), dim3(TPB), lds_bytes, stream>>>(in, out);
}